// Decoder_68083821576922
// MI455X (gfx1250) — compile-verified
//
#include <hip/hip_runtime.h>
#include <hip/hip_bf16.h>

#define N_NODES 10000
#define M_NB    32
#define DIM     256
#define FAN_IN  512   // NODE_DIM + EDGE_DIM
#define TILE    128
#define KC      32
#define LDS_K   (KC + 4)   // stride 36 words: 16B-aligned rows + conflict-free frag reads
#define NCHUNK  (DIM / KC) // 8

typedef __attribute__((ext_vector_type(2))) float v2f;
typedef __attribute__((ext_vector_type(8))) float v8f;
typedef __attribute__((ext_vector_type(4))) int   v4i;

typedef __attribute__((address_space(1))) v4i   as1_v4i;
typedef __attribute__((address_space(3))) v4i   as3_v4i;
typedef __attribute__((address_space(3))) float as3_f;

__device__ __forceinline__ float wave_reduce_add(float v) {
  #pragma unroll
  for (int off = 16; off > 0; off >>= 1) v += __shfl_xor(v, off, 32);
  return v;
}
__device__ __forceinline__ float wave_reduce_max(float v) {
  #pragma unroll
  for (int off = 16; off > 0; off >>= 1) v = fmaxf(v, __shfl_xor(v, off, 32));
  return v;
}
__device__ __forceinline__ float sigmoidf_fast(float x) {
  return 1.0f / (1.0f + __expf(-x));
}

// 16-byte async copy global -> LDS (no VGPR round-trip, tracked by ASYNCcnt).
__device__ __forceinline__ void async_copy16(const float* gsrc, float* ldst) {
#if __has_builtin(__builtin_amdgcn_global_load_async_to_lds_b128)
  __builtin_amdgcn_global_load_async_to_lds_b128(
      (as1_v4i*)gsrc, (as3_v4i*)ldst, 0, 0);
#else
  unsigned loff = (unsigned)(__UINTPTR_TYPE__)(as3_f*)ldst;
  asm volatile("global_load_async_to_lds_b128 %0, %1, off"
               :: "v"(loff), "v"(gsrc) : "memory");
#endif
}
__device__ __forceinline__ void wait_async0() {
#if __has_builtin(__builtin_amdgcn_s_wait_asynccnt)
  __builtin_amdgcn_s_wait_asynccnt(0);
#else
  asm volatile("s_wait_asynccnt 0x0" ::: "memory");
#endif
}

// ---------------------------------------------------------------------------
// Kernel A: per-node attention + context + attribute decoder (fused).
// ---------------------------------------------------------------------------
__global__ __launch_bounds__(256)
void attn_decode_kernel(const float* __restrict__ node_feature,
                        const float* __restrict__ agg,
                        const int*   __restrict__ nb_id,
                        const float* __restrict__ W,
                        const float* __restrict__ b,
                        float*       __restrict__ out_attr) {
  __shared__ float sA[DIM];
  __shared__ float sNB[M_NB][DIM];
  __shared__ float sH[FAN_IN];
  __shared__ float sScore[M_NB];
  __shared__ float sAttn[M_NB];
  __shared__ int   sIdx[M_NB];

  const int n    = blockIdx.x;
  const int t    = threadIdx.x;
  const int lane = t & 31;
  const int w    = t >> 5;

  sA[t] = agg[(size_t)n * DIM + t];
  sH[t] = node_feature[(size_t)n * DIM + t];
  if (t < M_NB) sIdx[t] = nb_id[n * M_NB + t];
  __syncthreads();

  for (int m = 0; m < M_NB; ++m) {
    int idx = sIdx[m];
    sNB[m][t] = (idx < 0) ? 0.0f : agg[(size_t)idx * DIM + t];
  }
  __syncthreads();

  #pragma unroll
  for (int q = 0; q < 4; ++q) {
    int m = w * 4 + q;
    float p = 0.0f;
    #pragma unroll
    for (int i = 0; i < 8; ++i)
      p += sA[lane + 32 * i] * sNB[m][lane + 32 * i];
    p = wave_reduce_add(p);
    if (lane == 0) sScore[m] = (sIdx[m] == -1) ? -1e10f : p;  // mask BEFORE scaling
  }
  __syncthreads();

  if (t < 32) {
    float s  = sScore[t] * (1.0f / 16.0f);      // 1/sqrt(256)
    float mx = wave_reduce_max(s);
    float e  = __expf(s - mx);
    float sm = wave_reduce_add(e);
    sAttn[t] = e / sm;
  }
  __syncthreads();

  float c = 0.0f;
  #pragma unroll 8
  for (int m = 0; m < M_NB; ++m) c += sAttn[m] * sNB[m][t];
  sH[DIM + t] = c;
  __syncthreads();

  float acc = b[t];
  #pragma unroll 8
  for (int k = 0; k < FAN_IN; ++k) acc += sH[k] * W[k * DIM + t];
  out_attr[(size_t)n * DIM + t] = sigmoidf_fast(acc);
}

// ---------------------------------------------------------------------------
// Kernel B: decode_adj = sigmoid(agg @ agg^T) via V_WMMA_F32_16X16X4_F32,
// with double-buffered GLOBAL_LOAD_ASYNC_TO_LDS_B128 panel staging.
// ---------------------------------------------------------------------------
__global__ __launch_bounds__(256)
void adj_wmma_kernel(const float* __restrict__ agg,
                     float*       __restrict__ out_adj) {
  __shared__ __align__(16) float As[2][TILE][LDS_K];  // 2 x 18 KB
  __shared__ __align__(16) float Bs[2][TILE][LDS_K];  // 2 x 18 KB

  const int t    = threadIdx.x;
  const int lane = t & 31;
  const int w    = t >> 5;            // wave id 0..7 -> row subtile
  const int half = lane >> 4;         // selects K pair / M half
  const int rsub = lane & 15;
  const int i0   = blockIdx.y * TILE;
  const int j0   = blockIdx.x * TILE;

  v8f acc[8] = {};

  // Async staging decomposition: 8 lanes x 16B cover one 128B row-chunk.
  const int seg = t & 7;              // 16B segment within row
  const int r0  = t >> 3;             // 0..31, 4 passes cover 128 rows

  // ---- prologue: stage chunk 0 into buffer 0 ----
  #pragma unroll
  for (int p = 0; p < 4; ++p) {
    int row = r0 + p * 32;
    int ga  = min(i0 + row, N_NODES - 1);
    int gb  = min(j0 + row, N_NODES - 1);
    async_copy16(agg + (size_t)ga * DIM + seg * 4, &As[0][row][seg * 4]);
    async_copy16(agg + (size_t)gb * DIM + seg * 4, &Bs[0][row][seg * 4]);
  }

  for (int c = 0; c < NCHUNK; ++c) {
    const int cur = c & 1;
    wait_async0();        // my wave's copies for chunk c complete
    __syncthreads();      // all waves' copies complete; prev compute on other buf done

    if (c + 1 < NCHUNK) { // stage next chunk into the other buffer (no wait)
      const int kn  = (c + 1) * KC;
      const int nxt = cur ^ 1;
      #pragma unroll
      for (int p = 0; p < 4; ++p) {
        int row = r0 + p * 32;
        int ga  = min(i0 + row, N_NODES - 1);
        int gb  = min(j0 + row, N_NODES - 1);
        async_copy16(agg + (size_t)ga * DIM + kn + seg * 4, &As[nxt][row][seg * 4]);
        async_copy16(agg + (size_t)gb * DIM + kn + seg * 4, &Bs[nxt][row][seg * 4]);
      }
    }

    // ---- compute on current buffer: 64 WMMAs, EXEC all-ones throughout ----
    #pragma unroll
    for (int ks = 0; ks < KC; ks += 4) {
      const int ka = ks + 2 * half;   // lane's K pair: frag[v] holds k = ka+v
      v2f a;
      a[0] = As[cur][w * 16 + rsub][ka + 0];
      a[1] = As[cur][w * 16 + rsub][ka + 1];
      #pragma unroll
      for (int j = 0; j < 8; ++j) {
        v2f bm;                        // B = agg^T -> same row-major addressing
        bm[0] = Bs[cur][j * 16 + rsub][ka + 0];
        bm[1] = Bs[cur][j * 16 + rsub][ka + 1];
        acc[j] = __builtin_amdgcn_wmma_f32_16x16x4_f32(
            false, a, false, bm, (short)0, acc[j], false, false);
      }
    }
    __syncthreads();      // compute on 'cur' done before it is restaged next round
  }

  // Store: D lane layout (M = v + 8*half, N = rsub); guards only after all WMMA.
  #pragma unroll
  for (int j = 0; j < 8; ++j) {
    int col = j0 + j * 16 + rsub;
    if (col < N_NODES) {
      #pragma unroll
      for (int v = 0; v < 8; ++v) {
        int row = i0 + w * 16 + v + 8 * half;
        if (row < N_NODES)
          out_adj[(size_t)row * N_NODES + col] = sigmoidf_fast(acc[j][v]);
      }
    }
  }
}

// ---------------------------------------------------------------------------
extern "C" void kernel_launch(void* const* d_in, const int* in_sizes, int n_in,
                              void* d_out, int out_size, void* d_ws, size_t ws_size,
                              hipStream_t stream) {
  const float* node_feature = (const float*)d_in[0];
  const float* agg          = (const float*)d_in[1];
  const int*   nb_id        = (const int*)d_in[2];
  const float* W            = (const float*)d_in[3];
  const float* b            = (const float*)d_in[4];

  float* out_attr = (float*)d_out;                            // [N, 256]
  float* out_adj  = (float*)d_out + (size_t)N_NODES * DIM;    // [N, N]

  attn_decode_kernel<<<N_NODES, 256, 0, stream>>>(node_feature, agg, nb_id, W, b,
                                                  out_attr);

  dim3 grid((N_NODES + TILE - 1) / TILE, (N_NODES + TILE - 1) / TILE);
  adj_wmma_kernel<<<grid, 256, 0, stream>>>(agg, out_adj);
}